// GCN_27178553049560
// MI455X (gfx1250) — compile-verified
//
#include <hip/hip_runtime.h>
#include <hip/hip_bf16.h>

// ---- problem constants (match reference) ----
#define N_NODES 100000
#define IN_F    256
#define HID     128
#define OUTF    64

typedef __attribute__((ext_vector_type(2))) float v2f;
typedef __attribute__((ext_vector_type(8))) float v8f;
typedef unsigned int v4ui __attribute__((ext_vector_type(4)));
typedef int v8i __attribute__((ext_vector_type(8)));
typedef int v4i __attribute__((ext_vector_type(4)));

// ---------------------------------------------------------------------------
// TDM: DMA a 16-row x K-float row-major strip (row stride = K floats in
// global) into LDS offset 0, inserting a 4-DWORD pad after every K DWORDs so
// the LDS row stride becomes K+4 floats (bank-conflict-free fragment reads:
// (K+4)%64 == 4 for K=128/256, so the 16 M-lanes land on distinct banks).
//
// Descriptor per CDNA5 ISA 8.3/8.4:
//  g0: [1:0]count=1 | [63:32]lds_addr | [120:64]global_addr | [127:126]type=2
//  g1: [17:16]data_size=2(4B) | [20]pad_enable | [24:22]pad_interval
//      [31:25]pad_amount | [79:48]tensor_dim0=K | [111:80]tensor_dim1=16
//      [127:112]tile_dim0=K | [143:128]tile_dim1=16 | [207:160]dim0_stride=K
//  remaining groups: zero (2D tensor, no gather/iterate)
//
// This toolchain exposes the 6-arg builtin:
//  (uint32x4 g0, int32x8 g1, int32x4, int32x4, int32x8, i32 cpol)
// ---------------------------------------------------------------------------
template <int K>
__device__ __forceinline__ void tdm_load_strip16(const float* gsrc)
{
    static_assert(K == 128 || K == 256, "pad interval encoding assumes K=128/256");
    constexpr unsigned IVCODE = (K == 256) ? 7u : 6u; // pad every 256 / 128 DWORDs
    constexpr unsigned PADAMT = 3u;                   // 4 DWORDs of pad (code n -> n+1)

    const unsigned long long ga = (unsigned long long)(uintptr_t)gsrc;

    v4ui g0;
    g0.x = 1u;                                        // count=1, user-mode load
    g0.y = 0u;                                        // lds_addr = 0 (single LDS buffer)
    g0.z = (unsigned)(ga & 0xFFFFFFFFu);              // global_addr[31:0]
    g0.w = (unsigned)((ga >> 32) & 0x01FFFFFFu)       // global_addr[56:32]
         | (2u << 30);                                // type = 2 ("image")

    v8i g1;
    g1[0] = (int)((2u << 16) | (1u << 20) | (IVCODE << 22) | (PADAMT << 25));
    g1[1] = (int)((unsigned)K << 16);                 // tensor_dim0[15:0] (K < 65536)
    g1[2] = (int)(16u << 16);                         // dim0[31:16]=0 | tensor_dim1=16
    g1[3] = (int)((unsigned)K << 16);                 // dim1[31:16]=0 | tile_dim0=K
    g1[4] = 16;                                       // tile_dim1=16 | tile_dim2=0
    g1[5] = K;                                        // tensor_dim0_stride[31:0]
    g1[6] = 0;                                        // stride hi | dim1_stride lo
    g1[7] = 0;

    const v4i z4 = {0, 0, 0, 0};
    const v8i z8 = {0, 0, 0, 0, 0, 0, 0, 0};
    __builtin_amdgcn_tensor_load_to_lds(g0, g1, z4, z4, z8, 0 /*cpol*/);
}

// ---------------------------------------------------------------------------
// Dense GEMM  C[M x N] = op(A)[M x K] @ B[K x N]   via V_WMMA_F32_16X16X4_F32
// op(A) = RELU_BIAS ? relu(A + bias[k]) : A   (fuses "relu(h + b1)" into the
// second GEMM's A-fragment load).
//
// One wave computes one 16x16 tile; A-strip staged in LDS by the TDM.
// blockDim.x = (N/16)*32, gridDim.x = M/16.  M multiple of 16 (100000 =
// 6250*16) so EXEC is all-ones for WMMA, as the ISA requires.
//
// f32 WMMA fragment layout (ISA 7.12.2):
//   A 16x4 : lanes 0-15 -> M=lane, VGPR0/1 = K0/K1 ; lanes 16-31 -> K2/K3
//   B 4x16 : mirrored (lanes 0-15 rows K0/K1, lanes 16-31 rows K2/K3)
//   C 16x16: 8 VGPRs, lane<16 -> M=r, lane>=16 -> M=r+8, N = lane&15
// ---------------------------------------------------------------------------
template <int K, int N, bool RELU_BIAS>
__global__ __launch_bounds__(256) void wmma_gemm_f32(
    const float* __restrict__ A, const float* __restrict__ B,
    const float* __restrict__ bias, float* __restrict__ C)
{
    constexpr int LDS_STRIDE = K + 4;                 // padded row stride (floats)
    __shared__ float As[16 * LDS_STRIDE];

    const int lane = threadIdx.x & 31;
    const int wave = threadIdx.x >> 5;
    const int m0   = blockIdx.x << 4;     // 16-row strip
    const int n0   = wave << 4;           // 16-col tile per wave
    const int mr   = lane & 15;
    const int kh   = (lane >> 4) << 1;    // 0 (lanes 0-15) or 2 (lanes 16-31)

    // Wave 0 issues the tensor DMA (scalar op, EXEC-independent), waits on its
    // own TENSORcnt, then the workgroup barrier publishes LDS to all waves.
    if (wave == 0) {
        tdm_load_strip16<K>(A + (size_t)m0 * K);
        __builtin_amdgcn_s_wait_tensorcnt(0);
    }
    __syncthreads();

    v8f acc = {};

    const float* ap = &As[mr * LDS_STRIDE + kh];            // LDS (ds_load_b64)
    const float* bp = B + (size_t)kh * N + n0 + mr;         // global

    for (int k0 = 0; k0 < K; k0 += 4) {
        v2f a, b;
        if (RELU_BIAS) {
            const float b0 = bias[k0 + kh];
            const float b1 = bias[k0 + kh + 1];
            a.x = fmaxf(ap[0] + b0, 0.0f);
            a.y = fmaxf(ap[1] + b1, 0.0f);
        } else {
            a.x = ap[0];
            a.y = ap[1];
        }
        b.x = bp[0];
        b.y = bp[N];
        // 8 args: (neg_a, A, neg_b, B, c_mod, C, reuse_a, reuse_b)
        acc = __builtin_amdgcn_wmma_f32_16x16x4_f32(
            false, a, false, b, (short)0, acc, false, false);
        ap += 4;
        bp += 4 * N;
    }

    const int   mo = m0 + ((lane >> 4) << 3);               // +8 for upper half
    float*      cp = C + (size_t)mo * N + n0 + (lane & 15);
#pragma unroll
    for (int r = 0; r < 8; ++r)
        cp[(size_t)r * N] = acc[r];
}

// ---------------------------------------------------------------------------
// Edge-parallel SpMM:  out[row[e], :] += vals[e] * H[col[e], :]
// One wave per edge; each lane handles F/32 contiguous features.
// Accumulation via global_atomic_add_f32 (edges are unsorted); operands are
// L2-resident (51MB / 25MB << 192MB L2).
// ---------------------------------------------------------------------------
template <int F>
__global__ __launch_bounds__(256) void spmm_atomic(
    const int* __restrict__ row, const int* __restrict__ col,
    const float* __restrict__ vals, const float* __restrict__ H,
    float* __restrict__ out, int nE)
{
    constexpr int FPL = F / 32; // floats per lane
    const int lane = threadIdx.x & 31;
    const int e    = blockIdx.x * (blockDim.x >> 5) + (threadIdx.x >> 5);
    if (e >= nE) return;

    const int   r = row[e];
    const int   c = col[e];
    const float v = vals[e];

    const float* hp = H + (size_t)c * F + lane * FPL;
    float*       op = out + (size_t)r * F + lane * FPL;

    float h[FPL];
#pragma unroll
    for (int i = 0; i < FPL; ++i) h[i] = hp[i];
#pragma unroll
    for (int i = 0; i < FPL; ++i) atomicAdd(op + i, v * h[i]);
}

// ---------------------------------------------------------------------------
__global__ void zero_f32(float* __restrict__ p, size_t n)
{
    size_t i      = (size_t)blockIdx.x * blockDim.x + threadIdx.x;
    size_t stride = (size_t)gridDim.x * blockDim.x;
    for (; i < n; i += stride) p[i] = 0.0f;
}

// out[i] = bias[i & 63]  (OUTF = 64, power of two)
__global__ void init_bias_rows64(float* __restrict__ out,
                                 const float* __restrict__ bias, size_t n)
{
    size_t i      = (size_t)blockIdx.x * blockDim.x + threadIdx.x;
    size_t stride = (size_t)gridDim.x * blockDim.x;
    for (; i < n; i += stride) out[i] = bias[i & 63];
}

// ---------------------------------------------------------------------------
extern "C" void kernel_launch(void* const* d_in, const int* in_sizes, int n_in,
                              void* d_out, int out_size, void* d_ws, size_t ws_size,
                              hipStream_t stream)
{
    // setup_inputs() order: x, w1, b1, w2, b2, adj_row, adj_col, adj_vals
    const float* x     = (const float*)d_in[0];
    const float* w1    = (const float*)d_in[1];
    const float* b1    = (const float*)d_in[2];
    const float* w2    = (const float*)d_in[3];
    const float* b2    = (const float*)d_in[4];
    const int*   arow  = (const int*)d_in[5];
    const int*   acol  = (const int*)d_in[6];
    const float* avals = (const float*)d_in[7];
    float*       out   = (float*)d_out;
    const int    nE    = in_sizes[5];

    // workspace partition: h1 (51.2MB) | h1s (51.2MB) | h2 (25.6MB) = 128MB
    float* h1  = (float*)d_ws;
    float* h1s = h1  + (size_t)N_NODES * HID;
    float* h2  = h1s + (size_t)N_NODES * HID;

    // 1. zero SpMM1 accumulator
    zero_f32<<<2048, 256, 0, stream>>>(h1s, (size_t)N_NODES * HID);

    // 2. h1 = x @ w1      (6250 blocks x 8 waves; TDM->LDS A-strip, f32 WMMA)
    wmma_gemm_f32<IN_F, HID, false>
        <<<N_NODES / 16, (HID / 16) * 32, 0, stream>>>(x, w1, nullptr, h1);

    // 3. h1s = spmm(h1)   (wave per edge, 4 atomics/lane)
    spmm_atomic<HID><<<(nE + 7) / 8, 256, 0, stream>>>(arow, acol, avals, h1, h1s, nE);

    // 4. h2 = relu(h1s + b1) @ w2   (bias+relu fused into A-fragment read)
    wmma_gemm_f32<HID, OUTF, true>
        <<<N_NODES / 16, (OUTF / 16) * 32, 0, stream>>>(h1s, w2, b1, h2);

    // 5. out[i,:] = b2    (so SpMM2 atomics land on out = spmm + b2)
    init_bias_rows64<<<2048, 256, 0, stream>>>(out, b2, (size_t)N_NODES * OUTF);

    // 6. out += spmm(h2)  (wave per edge, 2 atomics/lane)
    spmm_atomic<OUTF><<<(nE + 7) / 8, 256, 0, stream>>>(arow, acol, avals, h2, out, nE);
}